// GCN_30485677867423
// MI455X (gfx1250) — compile-verified
//
#include <hip/hip_runtime.h>
#include <math.h>

typedef float v2f __attribute__((ext_vector_type(2)));
typedef float v8f __attribute__((ext_vector_type(8)));

#define NN 12288
#define EE 393216
#define HH 16

// ---------------------------------------------------------------------------
// init: agg1 = 0, agg2 = 0, deg = 1.0 (self-loop pre-counted)
// ---------------------------------------------------------------------------
__global__ __launch_bounds__(256) void init_kernel(float* __restrict__ deg,
                                                   float* __restrict__ agg1,
                                                   float* __restrict__ agg2) {
    int idx = blockIdx.x * 256 + threadIdx.x;          // 0 .. 16N-1
    agg1[idx] = 0.0f;
    agg2[idx] = 0.0f;
    if (idx < NN) deg[idx] = 1.0f;
}

// ---------------------------------------------------------------------------
// degree: deg[dst] += 1 per edge
// ---------------------------------------------------------------------------
__global__ __launch_bounds__(256) void degree_kernel(const int* __restrict__ edst,
                                                     float* __restrict__ deg) {
    int e = blockIdx.x * 256 + threadIdx.x;
    if (e < EE) atomicAdd(deg + edst[e], 1.0f);
}

// ---------------------------------------------------------------------------
// dinv = rsqrt(deg)
// ---------------------------------------------------------------------------
__global__ __launch_bounds__(256) void dinv_kernel(float* __restrict__ deg) {
    int i = blockIdx.x * 256 + threadIdx.x;
    if (i < NN) {
        float d = deg[i];
        deg[i] = d > 0.0f ? rsqrtf(d) : 0.0f;
    }
}

// ---------------------------------------------------------------------------
// GEMM1: h1[N,16] = x[N,N] @ W1[N,16]   (fp32 WMMA 16x16x4)
// 768 blocks, 8 waves each; block owns 16 rows, waves split K, LDS reduce.
// ---------------------------------------------------------------------------
__global__ __launch_bounds__(256) void gemm1_kernel(const float* __restrict__ x,
                                                    const float* __restrict__ W1,
                                                    float* __restrict__ h1) {
    __shared__ float red[8][16][16];
    const int tid  = threadIdx.x;
    const int lane = tid & 31;
    const int wave = tid >> 5;
    const int l15  = lane & 15;          // A: row of x tile ; B: col of W1
    const int hi   = lane >> 4;          // 0/1
    const int koff = hi << 1;            // 0 or 2 (A/B K split per ISA layout)
    const int mt   = blockIdx.x;

    const float* xrow = x + (size_t)(mt * 16 + l15) * NN;
    v8f c = {};
    const int k0 = wave * (NN / 8);      // 1536 K per wave
#pragma unroll 4
    for (int k = k0; k < k0 + (NN / 8); k += 4) {
        v2f a = *(const v2f*)(xrow + k + koff);                 // global_load_b64
        v2f b;
        b.x = W1[(k + koff) * HH + l15];
        b.y = W1[(k + koff + 1) * HH + l15];
        c = __builtin_amdgcn_wmma_f32_16x16x4_f32(false, a, false, b,
                                                  (short)0, c, false, false);
    }
#pragma unroll
    for (int r = 0; r < 8; ++r) red[wave][r + hi * 8][l15] = c[r];
    __syncthreads();
    const int M = tid >> 4, Nc = tid & 15;
    float s = 0.0f;
#pragma unroll
    for (int w = 0; w < 8; ++w) s += red[w][M][Nc];
    h1[(size_t)(mt * 16 + M) * HH + Nc] = s;
}

// ---------------------------------------------------------------------------
// aggregate: hout[dst] += hin[src] * dinv[src]*dinv[dst]; one thread per
// (edge, feature); virtual edges e in [E, E+N) are self-loops.
// ---------------------------------------------------------------------------
__global__ __launch_bounds__(256) void aggregate_kernel(const int* __restrict__ esrc,
                                                        const int* __restrict__ edst,
                                                        const float* __restrict__ dinv,
                                                        const float* __restrict__ hin,
                                                        float* __restrict__ hout) {
    long long gid = (long long)blockIdx.x * 256 + threadIdx.x;
    long long e = gid >> 4;
    int f = (int)(gid & 15);
    if (e >= (long long)EE + NN) return;
    int s, d;
    if (e < EE) { s = esrc[e]; d = edst[e]; }
    else        { s = d = (int)(e - EE); }
    float w = dinv[s] * dinv[d];
    atomicAdd(hout + (size_t)d * HH + f, hin[(size_t)s * HH + f] * w);
}

// ---------------------------------------------------------------------------
// bias + relu (in place):  a = max(a + b1[f], 0)
// ---------------------------------------------------------------------------
__global__ __launch_bounds__(256) void biasrelu_kernel(float* __restrict__ a,
                                                       const float* __restrict__ b1) {
    int idx = blockIdx.x * 256 + threadIdx.x;        // 16N threads
    float v = a[idx] + b1[idx & 15];
    a[idx] = v > 0.0f ? v : 0.0f;
}

// ---------------------------------------------------------------------------
// Pass A: per-row max + log-sum-exp of logits = h2 @ W2 + b2 (never stored).
// 768 blocks (16 rows each), waves stride over 768 column tiles.
// ---------------------------------------------------------------------------
__global__ __launch_bounds__(256) void rowstats_kernel(const float* __restrict__ h2,
                                                       const float* __restrict__ W2,
                                                       const float* __restrict__ b2,
                                                       float* __restrict__ lse) {
    __shared__ float sm[8][16];
    __shared__ float ss[8][16];
    const int tid  = threadIdx.x;
    const int lane = tid & 31;
    const int wave = tid >> 5;
    const int l15  = lane & 15;
    const int hi   = lane >> 4;
    const int koff = hi << 1;
    const int mt   = blockIdx.x;

    const float* arow = h2 + (size_t)(mt * 16 + l15) * HH;
    v2f a0 = *(const v2f*)(arow + 0 + koff);
    v2f a1 = *(const v2f*)(arow + 4 + koff);
    v2f a2 = *(const v2f*)(arow + 8 + koff);
    v2f a3 = *(const v2f*)(arow + 12 + koff);

    float rm[8], rs[8];
#pragma unroll
    for (int r = 0; r < 8; ++r) { rm[r] = -3.402823466e38f; rs[r] = 0.0f; }

    for (int ct = wave; ct < NN / 16; ct += 8) {
        const int col = ct * 16 + l15;
        v2f b0, b1v, b2v, b3v;
        b0.x  = W2[(0  + koff) * NN + col]; b0.y  = W2[(1  + koff) * NN + col];
        b1v.x = W2[(4  + koff) * NN + col]; b1v.y = W2[(5  + koff) * NN + col];
        b2v.x = W2[(8  + koff) * NN + col]; b2v.y = W2[(9  + koff) * NN + col];
        b3v.x = W2[(12 + koff) * NN + col]; b3v.y = W2[(13 + koff) * NN + col];
        v8f c = {};
        c = __builtin_amdgcn_wmma_f32_16x16x4_f32(false, a0, false, b0,  (short)0, c, false, false);
        c = __builtin_amdgcn_wmma_f32_16x16x4_f32(false, a1, false, b1v, (short)0, c, false, false);
        c = __builtin_amdgcn_wmma_f32_16x16x4_f32(false, a2, false, b2v, (short)0, c, false, false);
        c = __builtin_amdgcn_wmma_f32_16x16x4_f32(false, a3, false, b3v, (short)0, c, false, false);
        const float bias = b2[col];
#pragma unroll
        for (int r = 0; r < 8; ++r) {
            float v = c[r] + bias;
            // 16-lane row reductions (xor masks stay within each half-wave)
            float m = v;
            m = fmaxf(m, __shfl_xor(m, 1, 32));
            m = fmaxf(m, __shfl_xor(m, 2, 32));
            m = fmaxf(m, __shfl_xor(m, 4, 32));
            m = fmaxf(m, __shfl_xor(m, 8, 32));
            float e = __expf(v - m);
            e += __shfl_xor(e, 1, 32);
            e += __shfl_xor(e, 2, 32);
            e += __shfl_xor(e, 4, 32);
            e += __shfl_xor(e, 8, 32);
            float nm = fmaxf(rm[r], m);
            rs[r] = rs[r] * __expf(rm[r] - nm) + e * __expf(m - nm);
            rm[r] = nm;
        }
    }
    if (l15 == 0) {            // lanes 0 and 16 of each wave hold the stats
#pragma unroll
        for (int r = 0; r < 8; ++r) { sm[wave][r + hi * 8] = rm[r]; ss[wave][r + hi * 8] = rs[r]; }
    }
    __syncthreads();
    if (tid < 16) {
        float M = -3.402823466e38f;
#pragma unroll
        for (int w = 0; w < 8; ++w) M = fmaxf(M, sm[w][tid]);
        float S = 0.0f;
#pragma unroll
        for (int w = 0; w < 8; ++w) S += ss[w][tid] * __expf(sm[w][tid] - M);
        lse[mt * 16 + tid] = M + __logf(S);
    }
}

// ---------------------------------------------------------------------------
// Pass B: recompute logits tile, write out = logits - lse[row]
// ---------------------------------------------------------------------------
__global__ __launch_bounds__(256) void writeout_kernel(const float* __restrict__ h2,
                                                       const float* __restrict__ W2,
                                                       const float* __restrict__ b2,
                                                       const float* __restrict__ lse,
                                                       float* __restrict__ out) {
    const int tid  = threadIdx.x;
    const int lane = tid & 31;
    const int wave = tid >> 5;
    const int l15  = lane & 15;
    const int hi   = lane >> 4;
    const int koff = hi << 1;
    const int mt   = blockIdx.x;

    const float* arow = h2 + (size_t)(mt * 16 + l15) * HH;
    v2f a0 = *(const v2f*)(arow + 0 + koff);
    v2f a1 = *(const v2f*)(arow + 4 + koff);
    v2f a2 = *(const v2f*)(arow + 8 + koff);
    v2f a3 = *(const v2f*)(arow + 12 + koff);

    float lsev[8];
#pragma unroll
    for (int r = 0; r < 8; ++r) lsev[r] = lse[mt * 16 + r + hi * 8];

    for (int ct = wave; ct < NN / 16; ct += 8) {
        const int col = ct * 16 + l15;
        v2f b0, b1v, b2v, b3v;
        b0.x  = W2[(0  + koff) * NN + col]; b0.y  = W2[(1  + koff) * NN + col];
        b1v.x = W2[(4  + koff) * NN + col]; b1v.y = W2[(5  + koff) * NN + col];
        b2v.x = W2[(8  + koff) * NN + col]; b2v.y = W2[(9  + koff) * NN + col];
        b3v.x = W2[(12 + koff) * NN + col]; b3v.y = W2[(13 + koff) * NN + col];
        v8f c = {};
        c = __builtin_amdgcn_wmma_f32_16x16x4_f32(false, a0, false, b0,  (short)0, c, false, false);
        c = __builtin_amdgcn_wmma_f32_16x16x4_f32(false, a1, false, b1v, (short)0, c, false, false);
        c = __builtin_amdgcn_wmma_f32_16x16x4_f32(false, a2, false, b2v, (short)0, c, false, false);
        c = __builtin_amdgcn_wmma_f32_16x16x4_f32(false, a3, false, b3v, (short)0, c, false, false);
        const float bias = b2[col];
#pragma unroll
        for (int r = 0; r < 8; ++r) {
            const size_t row = (size_t)(mt * 16 + r + hi * 8);
            out[row * NN + col] = c[r] + bias - lsev[r];
        }
    }
}

// ---------------------------------------------------------------------------
extern "C" void kernel_launch(void* const* d_in, const int* in_sizes, int n_in,
                              void* d_out, int out_size, void* d_ws, size_t ws_size,
                              hipStream_t stream) {
    const float* x    = (const float*)d_in[0];
    const int*   eidx = (const int*)d_in[1];     // [2, E] flat
    const float* W1   = (const float*)d_in[2];
    const float* b1   = (const float*)d_in[3];
    const float* W2   = (const float*)d_in[4];
    const float* b2   = (const float*)d_in[5];
    float* out = (float*)d_out;

    const int* esrc = eidx;
    const int* edst = eidx + EE;

    float* wsf  = (float*)d_ws;
    float* dinv = wsf;               // N      (deg -> dinv in place)
    float* h1   = wsf + NN;          // 16N    x @ W1
    float* agg1 = wsf + 17 * NN;     // 16N    aggregate1 -> relu in place
    float* agg2 = wsf + 33 * NN;     // 16N    aggregate2 (= h2)
    float* lse  = wsf + 49 * NN;     // N      row log-sum-exp

    const int tilesM   = NN / 16;                          // 768
    const int edgeThr  = (EE + NN) * HH;                   // (E+N)*16
    const int edgeBlk  = (edgeThr + 255) / 256;

    init_kernel<<<(16 * NN) / 256, 256, 0, stream>>>(dinv, agg1, agg2);
    degree_kernel<<<(EE + 255) / 256, 256, 0, stream>>>(edst, dinv);
    dinv_kernel<<<(NN + 255) / 256, 256, 0, stream>>>(dinv);
    gemm1_kernel<<<tilesM, 256, 0, stream>>>(x, W1, h1);
    aggregate_kernel<<<edgeBlk, 256, 0, stream>>>(esrc, edst, dinv, h1, agg1);
    biasrelu_kernel<<<(16 * NN) / 256, 256, 0, stream>>>(agg1, b1);
    aggregate_kernel<<<edgeBlk, 256, 0, stream>>>(esrc, edst, dinv, agg1, agg2);
    rowstats_kernel<<<tilesM, 256, 0, stream>>>(agg2, W2, b2, lse);
    writeout_kernel<<<tilesM, 256, 0, stream>>>(agg2, W2, b2, lse, out);
}